// WassersteinApproximation_53111565582834
// MI455X (gfx1250) — compile-verified
//
#include <hip/hip_runtime.h>
#include <math.h>

typedef __attribute__((ext_vector_type(2))) float v2f;
typedef __attribute__((ext_vector_type(8))) float v8f;

#define NT   512           // 16 wave32 per block
#define NW   (NT / 32)     // 16 waves
#define NSZ  48            // image side
#define NP   49            // padded LDS row stride (bank-conflict avoidance)
#define NPIX (NSZ * NSZ)   // 2304
#define REGC 5.0f
#define DIVC 1e-8f
#define TOLSQ 1e-10f       // TOL^2 vs ||u_prev - u_new||^2
#define MAXIT 250

// Deterministic block-wide sum: intra-wave __shfl_xor butterfly (no barriers),
// one partial per wave to LDS, then every thread sums the 16 partials in a
// fixed order (broadcast LDS reads). 2 barriers total vs ~12 for a full tree.
__device__ __forceinline__ float blockSum(float v, float* red, int tid) {
#pragma unroll
  for (int off = 16; off > 0; off >>= 1) v += __shfl_xor(v, off, 32);
  __syncthreads();  // protect red[] from previous readers
  if ((tid & 31) == 0) red[tid >> 5] = v;
  __syncthreads();
  float r = 0.f;
#pragma unroll
  for (int w = 0; w < NW; ++w) r += red[w];
  return r;  // identical value & order on every thread -> deterministic
}

// dst = L * R (+ dst if accum). 48x48x48 f32 GEMM on the matrix pipe:
// 9 output tiles of 16x16, one wave each, K consumed in steps of 4 via
// V_WMMA_F32_16X16X4_F32. Operand layouts per CDNA5 ISA 7.12.2:
//   A 16x4 f32 : lanes 0-15 row M, v0=K0 v1=K1; lanes 16-31: v0=K2 v1=K3
//   B 4x16 f32 : v0 = rows {K0|K2} striped over lanes, v1 = rows {K1|K3}
//   C/D 16x16  : VGPR g, lane half h -> M = g + 8h, N = lane&15
__device__ __forceinline__ void gemm48(float (*dst)[NP], float (*L)[NP],
                                       float (*R)[NP], int tid, bool accum) {
  __syncthreads();
  const int wave = tid >> 5;
  if (wave < 9) {
    const int lane = tid & 31;
    const int tm = (wave / 3) * 16;
    const int tn = (wave % 3) * 16;
    const int hl = lane >> 4;
    const int lr = lane & 15;
    v8f acc = {0.f, 0.f, 0.f, 0.f, 0.f, 0.f, 0.f, 0.f};
    if (accum) {
#pragma unroll
      for (int g = 0; g < 8; ++g) acc[g] = dst[tm + 8 * hl + g][tn + lr];
    }
#pragma unroll
    for (int k0 = 0; k0 < NSZ; k0 += 4) {
      const int ka = k0 + 2 * hl;
      v2f afr, bfr;
      afr.x = L[tm + lr][ka];
      afr.y = L[tm + lr][ka + 1];
      bfr.x = R[ka][tn + lr];
      bfr.y = R[ka + 1][tn + lr];
      acc = __builtin_amdgcn_wmma_f32_16x16x4_f32(false, afr, false, bfr,
                                                  (short)0, acc, false, false);
    }
#pragma unroll
    for (int g = 0; g < 8; ++g) dst[tm + 8 * hl + g][tn + lr] = acc[g];
  }
  __syncthreads();
}

// One workgroup == one sample. Full Sinkhorn loop in LDS.
// u@K == K@u (K symmetric) == vec(A*U*A) + DIVC*sum(U)  (error < f32 eps).
__global__ __launch_bounds__(NT) void sinkhorn48(const float* __restrict__ x,
                                                 const float* __restrict__ y,
                                                 float* __restrict__ out) {
  __shared__ float sA[NSZ][NP];   // exp(-5|i-j|)
  __shared__ float sXB[NSZ][NP];  // xs during loop; Bm=|i-j|e^{-5|i-j|} at finale
  __shared__ float sYK[NSZ][NP];  // ys during loop; (K*C)@v at finale
  __shared__ float sU[NSZ][NP];
  __shared__ float sV[NSZ][NP];   // doubles as stage-2 GEMM destination
  __shared__ float sT1[NSZ][NP];  // stage-1 GEMM temp
  __shared__ float sRed[NW];
  __shared__ float sMr[NSZ], sMc[NSZ], sF[NSZ], sG[NSZ];

  const int tid = threadIdx.x;
  const float* xb = x + blockIdx.x * NPIX;
  const float* yb = y + blockIdx.x * NPIX;

  // Tables + raw load + image sums
  float px = 0.f, py = 0.f;
  for (int idx = tid; idx < NPIX; idx += NT) {
    const int i = idx / NSZ, j = idx % NSZ;
    sA[i][j] = expf(-REGC * fabsf((float)(i - j)));
    const float vx = xb[idx], vy = yb[idx];
    sXB[i][j] = vx;
    sYK[i][j] = vy;
    px += vx;
    py += vy;
  }
  const float isx = 1.f / blockSum(px, sRed, tid);
  const float isy = 1.f / blockSum(py, sRed, tid);
  for (int idx = tid; idx < NPIX; idx += NT) {
    const int i = idx / NSZ, j = idx % NSZ;
    sXB[i][j] *= isx;
    sYK[i][j] *= isy;
    sU[i][j] = 1.f / (float)NPIX;
  }

  // ---- Sinkhorn loop (early-exit == reference's per-sample freeze) ----
  for (int iter = 0; iter < MAXIT; ++iter) {
    float pu = 0.f;
    for (int idx = tid; idx < NPIX; idx += NT) pu += sU[idx / NSZ][idx % NSZ];
    const float dU = DIVC * blockSum(pu, sRed, tid);
    gemm48(sT1, sA, sU, tid, false);  // A*U
    gemm48(sV, sT1, sA, tid, false);  // (A*U)*A  -> into V buffer
    for (int idx = tid; idx < NPIX; idx += NT) {
      const int i = idx / NSZ, j = idx % NSZ;
      sV[i][j] = sYK[i][j] / (sV[i][j] + dU);  // v = ys / (u@K)
    }
    float pv = 0.f;
    for (int idx = tid; idx < NPIX; idx += NT) pv += sV[idx / NSZ][idx % NSZ];
    const float dV = DIVC * blockSum(pv, sRed, tid);
    gemm48(sT1, sA, sV, tid, false);  // A*V
    gemm48(sV, sT1, sA, tid, false);  // (A*V)*A  (V dead, reuse as temp)
    float pd = 0.f;
    for (int idx = tid; idx < NPIX; idx += NT) {
      const int i = idx / NSZ, j = idx % NSZ;
      const float un = sXB[i][j] / (sV[i][j] + dV);  // u_new = xs / (v@K)
      const float du = un - sU[i][j];
      pd += du * du;
      sU[i][j] = un;
    }
    const float dsq = blockSum(pd, sRed, tid);  // uniform -> uniform break
    if (iter > 0 && dsq < TOLSQ) break;
  }

  // ---- final v = ys / (u@K) ----
  {
    float pu = 0.f;
    for (int idx = tid; idx < NPIX; idx += NT) pu += sU[idx / NSZ][idx % NSZ];
    const float dU = DIVC * blockSum(pu, sRed, tid);
    gemm48(sT1, sA, sU, tid, false);
    gemm48(sV, sT1, sA, tid, false);
    for (int idx = tid; idx < NPIX; idx += NT) {
      const int i = idx / NSZ, j = idx % NSZ;
      sV[i][j] = sYK[i][j] / (sV[i][j] + dU);
    }
  }

  // ---- (K*C)@v = Bm*V*A + A*V*Bm + DIVC*(C@v - near-overcount) ----
  for (int idx = tid; idx < NPIX; idx += NT) {  // Bm into xs storage (xs dead)
    const int i = idx / NSZ, j = idx % NSZ;
    const float d = fabsf((float)(i - j));
    sXB[i][j] = d * expf(-REGC * d);
  }
  gemm48(sT1, sXB, sV, tid, false);  // Bm*V
  gemm48(sYK, sT1, sA, tid, false);  // (Bm*V)*A        (ys dead -> KCv buffer)
  gemm48(sT1, sA, sV, tid, false);   // A*V
  gemm48(sYK, sT1, sXB, tid, true);  // += (A*V)*Bm

  // DIVC * C@v via marginals: C@v[(r,c)] = f[r] + g[c]
  if (tid < NSZ) {
    float s = 0.f;
    for (int j = 0; j < NSZ; ++j) s += sV[tid][j];
    sMr[tid] = s;
  } else if (tid < 2 * NSZ) {
    const int c = tid - NSZ;
    float s = 0.f;
    for (int r = 0; r < NSZ; ++r) s += sV[r][c];
    sMc[c] = s;
  }
  __syncthreads();
  if (tid < NSZ) {
    float s = 0.f;
    for (int r = 0; r < NSZ; ++r) s += fabsf((float)(tid - r)) * sMr[r];
    sF[tid] = s;
  } else if (tid < 2 * NSZ) {
    const int c = tid - NSZ;
    float s = 0.f;
    for (int cc = 0; cc < NSZ; ++cc) s += fabsf((float)(c - cc)) * sMc[cc];
    sG[c] = s;
  }
  __syncthreads();

  // dist = sum_i u_i * KCv_i ; subtract DIVC*cost over the near (cost<=3)
  // diamond where the clamp does not apply.
  float pdist = 0.f;
  for (int idx = tid; idx < NPIX; idx += NT) {
    const int r = idx / NSZ, c = idx % NSZ;
    float near = 0.f;
#pragma unroll
    for (int dr = -3; dr <= 3; ++dr) {
      const int rr = r + dr;
      if (rr < 0 || rr >= NSZ) continue;
      const int adr = (dr < 0) ? -dr : dr;
      const int lim = 3 - adr;
      for (int dc = -lim; dc <= lim; ++dc) {
        const int cc = c + dc;
        if (cc < 0 || cc >= NSZ) continue;
        const int cst = adr + ((dc < 0) ? -dc : dc);
        if (cst == 0) continue;
        near += (float)cst * sV[rr][cc];
      }
    }
    const float kcv = sYK[r][c] + DIVC * (sF[r] + sG[c] - near);
    pdist += sU[r][c] * kcv;
  }
  const float dist = blockSum(pdist, sRed, tid);
  if (tid == 0) out[blockIdx.x] = dist;
}

extern "C" void kernel_launch(void* const* d_in, const int* in_sizes, int n_in,
                              void* d_out, int out_size, void* d_ws, size_t ws_size,
                              hipStream_t stream) {
  (void)n_in;
  (void)out_size;
  (void)d_ws;
  (void)ws_size;
  const float* x = (const float*)d_in[0];
  const float* y = (const float*)d_in[1];
  float* out = (float*)d_out;
  const int B = in_sizes[0] / NPIX;  // 8 samples, one workgroup each
  sinkhorn48<<<dim3(B), dim3(NT), 0, stream>>>(x, y, out);
}